// GATNet_8340826488982
// MI455X (gfx1250) — compile-verified
//
#include <hip/hip_runtime.h>
#include <math.h>

#define HID 128
#define HEADS 4
#define CDIM 32
#define NCONV 3
#define NCLS 10
#define EPS 1e-5f
#define NEG_SLOPE 0.2f

typedef __attribute__((ext_vector_type(2))) float v2f;
typedef __attribute__((ext_vector_type(8))) float v8f;

// ---------------------------------------------------------------------------
// BatchNorm statistics: column sums and sums of squares over `rows` rows.
// sums[0..127] = sum, sums[128..255] = sumsq. Must be zeroed beforehand.
// ---------------------------------------------------------------------------
__global__ void bn_stats_kernel(const float* __restrict__ X, int rows,
                                float* __restrict__ sums) {
    int col = threadIdx.x;              // 128 threads
    float s = 0.f, ss = 0.f;
    for (int r = blockIdx.x; r < rows; r += gridDim.x) {
        float v = X[r * HID + col];
        s += v;
        ss += v * v;
    }
    atomicAdd(&sums[col], s);
    atomicAdd(&sums[HID + col], ss);
}

__global__ void bn_finalize_kernel(const float* __restrict__ sums,
                                   const float* __restrict__ gamma,
                                   const float* __restrict__ beta,
                                   float rows, float* __restrict__ scale,
                                   float* __restrict__ shift) {
    int c = threadIdx.x;                // 128 threads
    float mean = sums[c] / rows;
    float var = sums[HID + c] / rows - mean * mean;   // biased, like jnp.var
    float a = gamma[c] * rsqrtf(var + EPS);
    scale[c] = a;
    shift[c] = beta[c] - mean * a;
}

// ---------------------------------------------------------------------------
// Fold BatchNorm of A into B and bias:  (A*s + t) @ B = A @ (diag(s)B) + t^T B
//   Bp[k][n]  = B[k][n] * scale[k]
//   biasp[n]  = (bias ? bias[n] : 0) + sum_k shift[k] * B[k][n]
// One block, 128 threads (one per output column).
// ---------------------------------------------------------------------------
__global__ void prep_B_kernel(const float* __restrict__ B,
                              const float* __restrict__ scale,
                              const float* __restrict__ shift,
                              const float* __restrict__ bias,
                              float* __restrict__ Bp,
                              float* __restrict__ biasp) {
    int n = threadIdx.x;                // 128 threads
    float acc = bias ? bias[n] : 0.f;
    for (int k = 0; k < HID; ++k) {
        float v = B[k * HID + n];
        Bp[k * HID + n] = v * scale[k];
        acc += shift[k] * v;
    }
    biasp[n] = acc;
}

// ---------------------------------------------------------------------------
// WMMA f32 GEMM: C[M,128] = act( A @ Bp[128,128] + biasp )
// 256 threads = 8 waves; wave w computes strip (blockIdx.x*8+w) of 16 rows.
// Bp staged in LDS pair-interleaved: Bs[(k>>1)*256 + n*2 + (k&1)], so a lane's
// (B[k][n], B[k+1][n]) fragment is one aligned ds_load_b64.
// M must be a multiple of 16 (50000 = 3125*16).
// ---------------------------------------------------------------------------
__global__ __launch_bounds__(256)
void gemm128_wmma_kernel(const float* __restrict__ A,
                         const float* __restrict__ Bp,
                         const float* __restrict__ biasp,
                         float* __restrict__ C, int M, int do_relu) {
    __shared__ float Bs[HID * HID];
    for (int i = threadIdx.x; i < HID * HID; i += 256) {
        int k = i >> 7;
        int n = i & (HID - 1);
        Bs[(k >> 1) * 256 + n * 2 + (k & 1)] = Bp[i];
    }
    __syncthreads();

    int wave = threadIdx.x >> 5;
    int lane = threadIdx.x & 31;
    int strip = blockIdx.x * 8 + wave;      // wave-uniform
    if (strip * 16 >= M) return;            // uniform branch: EXEC stays full
    int base = strip * 16;

    int m = lane & 15;
    int kh = lane >> 4;                     // lanes 0-15 -> K+0/1, 16-31 -> K+2/3
    const float* Arow = A + (size_t)(base + m) * HID + kh * 2;
    const float* Brow = Bs + kh * 256 + m * 2;   // + k0*32 + t*32 per fragment

    v8f zero = {0.f, 0.f, 0.f, 0.f, 0.f, 0.f, 0.f, 0.f};
    v8f acc[8];
#pragma unroll
    for (int t = 0; t < 8; ++t) acc[t] = zero;

    for (int k0 = 0; k0 < HID; k0 += 4) {
        v2f a = *(const v2f*)(Arow + k0);   // one 8-byte global load
        const float* bk = Brow + k0 * 32;   // k0*32 + kh*256 + m*2
#pragma unroll
        for (int t = 0; t < 8; ++t) {
            v2f b = *(const v2f*)(bk + t * 32);  // one aligned ds_load_b64
            acc[t] = __builtin_amdgcn_wmma_f32_16x16x4_f32(
                false, a, false, b, (short)0, acc[t], false, false);
        }
    }

    int rowoff = kh << 3;                   // lanes 16-31 hold rows M=8..15
#pragma unroll
    for (int t = 0; t < 8; ++t) {
        int col = t * 16 + m;
        float bv = biasp[col];
#pragma unroll
        for (int j = 0; j < 8; ++j) {
            float v = acc[t][j] + bv;
            if (do_relu) v = v > 0.f ? v : 0.f;
            C[(size_t)(base + rowoff + j) * HID + col] = v;
        }
    }
}

// ---------------------------------------------------------------------------
// Attention scores per node per head: a = sum_c hh[n,h,c]*att[h,c]
// ---------------------------------------------------------------------------
__global__ void att_kernel(const float* __restrict__ hh,
                           const float* __restrict__ att_s,
                           const float* __restrict__ att_d,
                           float* __restrict__ a_s, float* __restrict__ a_d,
                           int N) {
    int gid = blockIdx.x * blockDim.x + threadIdx.x;
    if (gid >= N * HEADS) return;
    int n = gid >> 2;
    int h = gid & 3;
    const float* row = hh + (size_t)n * HID + h * CDIM;
    const float* vs = att_s + h * CDIM;
    const float* vd = att_d + h * CDIM;
    float s = 0.f, d = 0.f;
#pragma unroll
    for (int c = 0; c < CDIM; ++c) {
        float v = row[c];
        s += v * vs[c];
        d += v * vd[c];
    }
    a_s[gid] = s;
    a_d[gid] = d;
}

__global__ void fill_kernel(float* __restrict__ p, float v, int n) {
    int i = blockIdx.x * blockDim.x + threadIdx.x;
    if (i < n) p[i] = v;
}

__device__ inline void atomicMaxFloat(float* addr, float val) {
    if (val >= 0.f)
        atomicMax((int*)addr, __float_as_int(val));
    else
        atomicMin((unsigned int*)addr, __float_as_uint(val));
}

__device__ inline void edge_src_tgt(const int* ei, int E, int idx, int& s, int& t) {
    if (idx < E) { s = ei[idx]; t = ei[E + idx]; }
    else         { s = t = idx - E; }   // self loops appended
}

// Pass 1: per-target per-head max of leaky_relu(a_s[src]+a_d[tgt])
__global__ void edge_max_kernel(const int* __restrict__ ei, int E, int N,
                                const float* __restrict__ a_s,
                                const float* __restrict__ a_d,
                                float* __restrict__ mmax) {
    int idx = blockIdx.x * blockDim.x + threadIdx.x;
    if (idx >= E + N) return;
    int s, t;
    edge_src_tgt(ei, E, idx, s, t);
#pragma unroll
    for (int h = 0; h < HEADS; ++h) {
        float e = a_s[s * HEADS + h] + a_d[t * HEADS + h];
        e = e > 0.f ? e : NEG_SLOPE * e;
        atomicMaxFloat(&mmax[t * HEADS + h], e);
    }
}

// Pass 2: ex = exp(e - max); store per-edge, accumulate denominator
__global__ void edge_exp_kernel(const int* __restrict__ ei, int E, int N,
                                const float* __restrict__ a_s,
                                const float* __restrict__ a_d,
                                const float* __restrict__ mmax,
                                float* __restrict__ exbuf,
                                float* __restrict__ denom) {
    int idx = blockIdx.x * blockDim.x + threadIdx.x;
    if (idx >= E + N) return;
    int s, t;
    edge_src_tgt(ei, E, idx, s, t);
#pragma unroll
    for (int h = 0; h < HEADS; ++h) {
        float e = a_s[s * HEADS + h] + a_d[t * HEADS + h];
        e = e > 0.f ? e : NEG_SLOPE * e;
        float ex = expf(e - mmax[t * HEADS + h]);
        exbuf[(size_t)idx * HEADS + h] = ex;
        atomicAdd(&denom[t * HEADS + h], ex);
    }
}

// Pass 3: out[tgt,c] += hh[src,c] * (ex/denom[tgt,head(c)])
__global__ void edge_agg_kernel(const int* __restrict__ ei, int E, int N,
                                const float* __restrict__ hh,
                                const float* __restrict__ exbuf,
                                const float* __restrict__ denom,
                                float* __restrict__ out) {
    int gid = blockIdx.x * blockDim.x + threadIdx.x;
    int idx = gid >> 7;
    int c = gid & (HID - 1);
    if (idx >= E + N) return;
    int s, t;
    edge_src_tgt(ei, E, idx, s, t);
    int h = c >> 5;
    float alpha = exbuf[(size_t)idx * HEADS + h] / denom[t * HEADS + h];
    atomicAdd(&out[(size_t)t * HID + c], hh[(size_t)s * HID + c] * alpha);
}

__global__ void relu_bias_kernel(const float* __restrict__ acc,
                                 const float* __restrict__ bias,
                                 float* __restrict__ h, int total) {
    int gid = blockIdx.x * blockDim.x + threadIdx.x;
    if (gid >= total) return;
    float v = acc[gid] + bias[gid & (HID - 1)];
    h[gid] = v > 0.f ? v : 0.f;
}

__global__ void pool_kernel(const float* __restrict__ h,
                            const int* __restrict__ batch,
                            float* __restrict__ g, int N) {
    int gid = blockIdx.x * blockDim.x + threadIdx.x;
    if (gid >= N * HID) return;
    int n = gid >> 7;
    int c = gid & (HID - 1);
    atomicAdd(&g[batch[n] * HID + c], h[gid]);
}

// Small batchnorm over `rows` rows (single block, 128 threads, per-column)
__global__ void bn_small_kernel(const float* __restrict__ X,
                                const float* __restrict__ gamma,
                                const float* __restrict__ beta,
                                float* __restrict__ Y, int rows) {
    int c = threadIdx.x;                // 128 threads
    float s = 0.f, ss = 0.f;
    for (int r = 0; r < rows; ++r) {
        float v = X[r * HID + c];
        s += v;
        ss += v * v;
    }
    float mean = s / rows;
    float var = ss / rows - mean * mean;
    float a = gamma[c] * rsqrtf(var + EPS);
    float b2 = beta[c] - mean * a;
    for (int r = 0; r < rows; ++r)
        Y[r * HID + c] = X[r * HID + c] * a + b2;
}

__global__ void fc_kernel(const float* __restrict__ X,
                          const float* __restrict__ W,
                          const float* __restrict__ bias,
                          float* __restrict__ Y, int rows, int ncols,
                          int do_relu) {
    int gid = blockIdx.x * blockDim.x + threadIdx.x;
    int r = gid / ncols;
    int c = gid - r * ncols;
    if (r >= rows) return;
    float acc = bias[c];
    for (int k = 0; k < HID; ++k) acc += X[r * HID + k] * W[k * ncols + c];
    if (do_relu) acc = acc > 0.f ? acc : 0.f;
    Y[r * ncols + c] = acc;
}

__global__ void logsoftmax_kernel(const float* __restrict__ L,
                                  float* __restrict__ out, int rows) {
    int r = blockIdx.x * blockDim.x + threadIdx.x;
    if (r >= rows) return;
    float l[NCLS];
    float mx = -INFINITY;
#pragma unroll
    for (int j = 0; j < NCLS; ++j) {
        l[j] = L[r * NCLS + j];
        mx = fmaxf(mx, l[j]);
    }
    float s = 0.f;
#pragma unroll
    for (int j = 0; j < NCLS; ++j) s += expf(l[j] - mx);
    float lse = mx + logf(s);
#pragma unroll
    for (int j = 0; j < NCLS; ++j) out[r * NCLS + j] = l[j] - lse;
}

// ---------------------------------------------------------------------------
extern "C" void kernel_launch(void* const* d_in, const int* in_sizes, int n_in,
                              void* d_out, int out_size, void* d_ws, size_t ws_size,
                              hipStream_t stream) {
    const float* x        = (const float*)d_in[0];
    const int*   ei       = (const int*)d_in[1];
    const int*   batch    = (const int*)d_in[2];
    const float* bnf_g    = (const float*)d_in[3];
    const float* bnf_b    = (const float*)d_in[4];
    const float* W0       = (const float*)d_in[5];
    const float* b0       = (const float*)d_in[6];
    const float* bn_g     = (const float*)d_in[7];
    const float* bn_b     = (const float*)d_in[8];
    const float* Wc       = (const float*)d_in[9];
    const float* att_src  = (const float*)d_in[10];
    const float* att_dst  = (const float*)d_in[11];
    const float* bc       = (const float*)d_in[12];
    const float* bn_fc_g  = (const float*)d_in[13];
    const float* bn_fc_b  = (const float*)d_in[14];
    const float* W_fc     = (const float*)d_in[15];
    const float* b_fc     = (const float*)d_in[16];
    const float* bn_h_g   = (const float*)d_in[17];
    const float* bn_h_b   = (const float*)d_in[18];
    const float* W_cls    = (const float*)d_in[19];
    const float* b_cls    = (const float*)d_in[20];

    const int N = in_sizes[0] / HID;
    const int E = in_sizes[1] / 2;
    const int TOT = E + N;
    const int G = out_size / NCLS;      // NGRAPH

    // Workspace layout (floats)
    float* w = (float*)d_ws;
    float* h      = w;                  w += (size_t)N * HID;
    float* hh     = w;                  w += (size_t)N * HID;
    float* oacc   = w;                  w += (size_t)N * HID;
    float* a_s    = w;                  w += (size_t)N * HEADS;
    float* a_d    = w;                  w += (size_t)N * HEADS;
    float* mmax   = w;                  w += (size_t)N * HEADS;
    float* denom  = w;                  w += (size_t)N * HEADS;
    float* exbuf  = w;                  w += (size_t)TOT * HEADS;
    float* sums   = w;                  w += 2 * HID;
    float* scale  = w;                  w += HID;
    float* shift  = w;                  w += HID;
    float* Bp     = w;                  w += HID * HID;
    float* biasp  = w;                  w += HID;
    float* g      = w;                  w += (size_t)G * HID;
    float* gbn    = w;                  w += (size_t)G * HID;
    float* g2     = w;                  w += (size_t)G * HID;
    float* gbn2   = w;                  w += (size_t)G * HID;
    float* logits = w;                  w += (size_t)G * NCLS;

    const int gemmGrid = (N / 16 + 7) / 8;          // 8 strips (waves) / block
    const int edgeGrid = (TOT + 255) / 256;
    const int aggGrid  = (TOT * HID + 255) / 256;
    const int nhGrid   = (N * HEADS + 255) / 256;
    const int ncGrid   = (N * HID + 255) / 256;

    // -------- input batchnorm + first linear (GCNConv gfn) + relu ----------
    hipMemsetAsync(sums, 0, 2 * HID * sizeof(float), stream);
    bn_stats_kernel<<<256, HID, 0, stream>>>(x, N, sums);
    bn_finalize_kernel<<<1, HID, 0, stream>>>(sums, bnf_g, bnf_b, (float)N, scale, shift);
    prep_B_kernel<<<1, HID, 0, stream>>>(W0, scale, shift, b0, Bp, biasp);
    gemm128_wmma_kernel<<<gemmGrid, 256, 0, stream>>>(x, Bp, biasp, h, N, 1);

    // ------------------------------ GAT layers -----------------------------
    for (int i = 0; i < NCONV; ++i) {
        hipMemsetAsync(sums, 0, 2 * HID * sizeof(float), stream);
        bn_stats_kernel<<<256, HID, 0, stream>>>(h, N, sums);
        bn_finalize_kernel<<<1, HID, 0, stream>>>(sums, bn_g + i * HID, bn_b + i * HID,
                                                  (float)N, scale, shift);
        // hh = BN(h) @ Wc[i]   (BN folded into Wc and bias)
        prep_B_kernel<<<1, HID, 0, stream>>>(Wc + (size_t)i * HID * HID, scale, shift,
                                             nullptr, Bp, biasp);
        gemm128_wmma_kernel<<<gemmGrid, 256, 0, stream>>>(h, Bp, biasp, hh, N, 0);

        att_kernel<<<nhGrid, 256, 0, stream>>>(hh, att_src + i * HID, att_dst + i * HID,
                                               a_s, a_d, N);

        fill_kernel<<<nhGrid, 256, 0, stream>>>(mmax, -INFINITY, N * HEADS);
        hipMemsetAsync(denom, 0, (size_t)N * HEADS * sizeof(float), stream);
        hipMemsetAsync(oacc, 0, (size_t)N * HID * sizeof(float), stream);

        edge_max_kernel<<<edgeGrid, 256, 0, stream>>>(ei, E, N, a_s, a_d, mmax);
        edge_exp_kernel<<<edgeGrid, 256, 0, stream>>>(ei, E, N, a_s, a_d, mmax,
                                                      exbuf, denom);
        edge_agg_kernel<<<aggGrid, 256, 0, stream>>>(ei, E, N, hh, exbuf, denom, oacc);

        relu_bias_kernel<<<ncGrid, 256, 0, stream>>>(oacc, bc + i * HID, h, N * HID);
    }

    // ------------------------ pooling + FC head ----------------------------
    hipMemsetAsync(g, 0, (size_t)G * HID * sizeof(float), stream);
    pool_kernel<<<ncGrid, 256, 0, stream>>>(h, batch, g, N);

    bn_small_kernel<<<1, HID, 0, stream>>>(g, bn_fc_g, bn_fc_b, gbn, G);
    fc_kernel<<<(G * HID + 255) / 256, 256, 0, stream>>>(gbn, W_fc, b_fc, g2, G, HID, 1);
    bn_small_kernel<<<1, HID, 0, stream>>>(g2, bn_h_g, bn_h_b, gbn2, G);
    fc_kernel<<<(G * NCLS + 255) / 256, 256, 0, stream>>>(gbn2, W_cls, b_cls, logits,
                                                          G, NCLS, 0);
    logsoftmax_kernel<<<(G + 255) / 256, 256, 0, stream>>>(logits, (float*)d_out, G);
}